// KeyCompressor_33071248179493
// MI455X (gfx1250) — compile-verified
//
#include <hip/hip_runtime.h>
#include <math.h>

// Problem constants (match reference)
#define R_ 8
#define H_ 8
#define K_ 4096
#define C_ 128
#define N_ 4096          // B*S = 2*2048
#define F_ 1024          // H*C

typedef __attribute__((ext_vector_type(2))) float v2f;
typedef __attribute__((ext_vector_type(8))) float v8f;

// ---------------------------------------------------------------------------
// Kernel 1: permute + normalize.  t0[n][h][c] = xp / ||xp||,  prescale[n]=||xp||
// xp[h*128+c] = x[h*128 + (c&1)*64 + (c>>1)]   (interleave the two 64-halves)
// ---------------------------------------------------------------------------
__global__ __launch_bounds__(256) void rvq_prep(const float* __restrict__ x,
                                                float* __restrict__ t0,
                                                float* __restrict__ prescale) {
  __shared__ float red[256];
  const int n = blockIdx.x;
  const int tid = threadIdx.x;
  const float* xrow = x + (size_t)n * F_;

  float vals[4];
  float ss = 0.f;
#pragma unroll
  for (int i = 0; i < 4; ++i) {
    int p = tid * 4 + i;               // permuted (output) position
    int h = p >> 7, c = p & 127;
    int src = h * 128 + ((c & 1) << 6) + (c >> 1);
    float v = xrow[src];
    vals[i] = v;
    ss += v * v;
  }
  red[tid] = ss;
  __syncthreads();
  for (int s = 128; s > 0; s >>= 1) {
    if (tid < s) red[tid] += red[tid + s];
    __syncthreads();
  }
  float nrm = sqrtf(red[0]);
  if (tid == 0) prescale[n] = nrm;
  float inv = 1.0f / nrm;
  float* trow = t0 + (size_t)n * F_;
#pragma unroll
  for (int i = 0; i < 4; ++i) trow[tid * 4 + i] = vals[i] * inv;
}

// ---------------------------------------------------------------------------
// Kernel 2: cnorm[r][h][k] = sum_c cb[r][h][k][c]^2
// ---------------------------------------------------------------------------
__global__ __launch_bounds__(256) void rvq_cnorm(const float* __restrict__ cb,
                                                 float* __restrict__ cnorm) {
  size_t i = (size_t)blockIdx.x * 256 + threadIdx.x;   // < R*H*K = 262144
  const v2f* row = (const v2f*)(cb + i * (size_t)C_);
  float s = 0.f;
#pragma unroll
  for (int j = 0; j < 64; ++j) {
    v2f v = row[j];
    s += v.x * v.x + v.y * v.y;
  }
  cnorm[i] = s;
}

// ---------------------------------------------------------------------------
// Kernel 3: fused 8-stage residual VQ + reconstruction.
// One wave owns (head h, 32 tokens = two 16x16 A-tiles).  Per 16-code k-tile:
// 64 x V_WMMA_F32_16X16X4_F32 across 4 independent accumulator chains
// (2 tiles x even/odd C-chunk), each B fragment reused by both tiles.
//
// Fragment layouts (CDNA5 ISA 7.12.2, 32-bit, wave32):
//   A 16x4 : lane L -> row M = L&15 ; VGPR{0,1} = K = (L>>4)*2 + {0,1}
//   B 4x16 : lane L -> col N = L&15 ; VGPR{0,1} = K = (L>>4)*2 + {0,1}
//   C 16x16: lane L, VGPR v -> row M = v + 8*(L>>4), col N = L&15
// ---------------------------------------------------------------------------
__global__ __launch_bounds__(128) void rvq_main(const float* __restrict__ cb,
                                                const float* __restrict__ t0g,
                                                const float* __restrict__ prescale,
                                                const float* __restrict__ cnorm,
                                                float* __restrict__ out) {
  __shared__ int s_idx[4][32];

  const int lane  = threadIdx.x & 31;
  const int wave  = threadIdx.x >> 5;
  const int h     = blockIdx.y;
  const int m     = lane & 15;      // token row (A) / code col (B,acc)
  const int khalf = lane >> 4;      // selects K-pair {0,1} vs {2,3}
  const int base  = (blockIdx.x * 4 + wave) * 32;
  const int tok0  = base + m;
  const int tok1  = base + 16 + m;

  // Load A fragments (residual t) for both token tiles: 128 VGPRs.
  v2f aT0[32], aT1[32];
  {
    const float* tr0 = t0g + (size_t)tok0 * F_ + h * C_ + khalf * 2;
    const float* tr1 = t0g + (size_t)tok1 * F_ + h * C_ + khalf * 2;
#pragma unroll
    for (int ch = 0; ch < 32; ++ch) {
      aT0[ch] = *(const v2f*)(tr0 + ch * 4);
      aT1[ch] = *(const v2f*)(tr1 + ch * 4);
    }
  }

  for (int r = 0; r < R_; ++r) {
    const float* cbr = cb + (((size_t)r * H_ + h) * K_) * C_;
    const float* cn  = cnorm + ((size_t)r * H_ + h) * K_;

    float minv0[8], minv1[8];
    int   mink0[8], mink1[8];
#pragma unroll
    for (int v = 0; v < 8; ++v) {
      minv0[v] = 3.402823e38f; mink0[v] = 0;
      minv1[v] = 3.402823e38f; mink1[v] = 0;
    }

    for (int kt = 0; kt < K_ / 16; ++kt) {
      const int krow = kt * 16 + m;                       // this lane's code
      const float* brow = cbr + (size_t)krow * C_ + khalf * 2;
      const float cn_k = cn[krow];

      v8f acc0[2] = {{}, {}};
      v8f acc1[2] = {{}, {}};
#pragma unroll
      for (int ch = 0; ch < 32; ++ch) {
        v2f b = *(const v2f*)(brow + ch * 4);
        acc0[ch & 1] = __builtin_amdgcn_wmma_f32_16x16x4_f32(
            false, aT0[ch], false, b, (short)0, acc0[ch & 1], false, false);
        acc1[ch & 1] = __builtin_amdgcn_wmma_f32_16x16x4_f32(
            false, aT1[ch], false, b, (short)0, acc1[ch & 1], false, false);
      }
#pragma unroll
      for (int v = 0; v < 8; ++v) {
        float s0 = cn_k - 2.0f * (acc0[0][v] + acc0[1][v]);
        float s1 = cn_k - 2.0f * (acc1[0][v] + acc1[1][v]);
        if (s0 < minv0[v]) { minv0[v] = s0; mink0[v] = krow; }
        if (s1 < minv1[v]) { minv1[v] = s1; mink1[v] = krow; }
      }
    }

    // Argmin across the 16 lanes of each half (codes interleave mod 16).
#pragma unroll
    for (int off = 1; off < 16; off <<= 1) {
#pragma unroll
      for (int v = 0; v < 8; ++v) {
        float ov0 = __shfl_xor(minv0[v], off, 32);
        int   ok0 = __shfl_xor(mink0[v], off, 32);
        if (ov0 < minv0[v] || (ov0 == minv0[v] && ok0 < mink0[v])) {
          minv0[v] = ov0; mink0[v] = ok0;
        }
        float ov1 = __shfl_xor(minv1[v], off, 32);
        int   ok1 = __shfl_xor(mink1[v], off, 32);
        if (ov1 < minv1[v] || (ov1 == minv1[v] && ok1 < mink1[v])) {
          minv1[v] = ov1; mink1[v] = ok1;
        }
      }
    }
    // lanes 0-15 hold tokens 0-7 (slot v), lanes 16-31 hold tokens 8-15.
    if (lane == 0) {
#pragma unroll
      for (int v = 0; v < 8; ++v) {
        s_idx[wave][v]      = mink0[v];
        s_idx[wave][16 + v] = mink1[v];
      }
    }
    if (lane == 16) {
#pragma unroll
      for (int v = 0; v < 8; ++v) {
        s_idx[wave][8 + v]  = mink0[v];
        s_idx[wave][24 + v] = mink1[v];
      }
    }
    __syncthreads();

    // Residual update: t -= cb[code]; each lane gathers its token's code row.
    const int myk0 = s_idx[wave][m];
    const int myk1 = s_idx[wave][16 + m];
    const float* q0row = cbr + (size_t)myk0 * C_ + khalf * 2;
    const float* q1row = cbr + (size_t)myk1 * C_ + khalf * 2;
#pragma unroll
    for (int ch = 0; ch < 32; ++ch) {
      v2f q0 = *(const v2f*)(q0row + ch * 4);
      v2f q1 = *(const v2f*)(q1row + ch * 4);
      aT0[ch].x -= q0.x; aT0[ch].y -= q0.y;
      aT1[ch].x -= q1.x; aT1[ch].y -= q1.y;
    }
    __syncthreads();
  }

  // Reconstruction: out = (t0 - t_final) * prescale  (== sum of selected codes)
  {
    const float ps0 = prescale[tok0];
    const float ps1 = prescale[tok1];
    const float* tr0 = t0g + (size_t)tok0 * F_ + h * C_ + khalf * 2;
    const float* tr1 = t0g + (size_t)tok1 * F_ + h * C_ + khalf * 2;
    float* o0 = out + (size_t)tok0 * F_ + h * C_ + khalf * 2;
    float* o1 = out + (size_t)tok1 * F_ + h * C_ + khalf * 2;
#pragma unroll
    for (int ch = 0; ch < 32; ++ch) {
      v2f t0v = *(const v2f*)(tr0 + ch * 4);
      v2f t1v = *(const v2f*)(tr1 + ch * 4);
      v2f w0, w1;
      w0.x = (t0v.x - aT0[ch].x) * ps0;
      w0.y = (t0v.y - aT0[ch].y) * ps0;
      w1.x = (t1v.x - aT1[ch].x) * ps1;
      w1.y = (t1v.y - aT1[ch].y) * ps1;
      *(v2f*)(o0 + ch * 4) = w0;
      *(v2f*)(o1 + ch * 4) = w1;
    }
  }
}

// ---------------------------------------------------------------------------
extern "C" void kernel_launch(void* const* d_in, const int* in_sizes, int n_in,
                              void* d_out, int out_size, void* d_ws, size_t ws_size,
                              hipStream_t stream) {
  const float* x  = (const float*)d_in[0];   // (B,S,H*C)  = 4,194,304 f32
  const float* cb = (const float*)d_in[1];   // (R,H,K,C)  = 33,554,432 f32
  float* out = (float*)d_out;                // (B,S,H*C) f32

  float* t0       = (float*)d_ws;                       // N*F floats (16 MB)
  float* prescale = t0 + (size_t)N_ * F_;               // N floats
  float* cnorm    = prescale + N_;                      // R*H*K floats (1 MB)

  rvq_prep<<<N_, 256, 0, stream>>>(x, t0, prescale);
  rvq_cnorm<<<(R_ * H_ * K_) / 256, 256, 0, stream>>>(cb, cnorm);

  dim3 grid(N_ / 128, H_);   // 4 waves/block * 32 tokens/wave
  rvq_main<<<grid, 128, 0, stream>>>(cb, t0, prescale, cnorm, out);
}